// MultiPerspective_14328010899892
// MI455X (gfx1250) — compile-verified
//
#include <hip/hip_runtime.h>
#include <math.h>

// MI455X / gfx1250, wave32. FP32 WMMA path (V_WMMA_F32_16X16X4_F32) to match
// the fp32 reference while using the matrix pipe. LDS staging uses the 320KB
// WGP LDS (dynamic shared memory, up to ~254KB per workgroup).

#define T_DIM 64
#define B_DIM 128
#define H_DIM 384
#define MP_DIM 20
#define EPS_F 1e-8f
#define XSTRIDE 388   // padded row stride: 388 % 64 banks = 4 -> conflict-free columns
#define CSTRIDE 68    // cos matrix LDS stride

typedef float v2f __attribute__((ext_vector_type(2)));
typedef float v8f __attribute__((ext_vector_type(8)));

__device__ __forceinline__ v8f wmma_f32_4(v2f a, v2f b, v8f c) {
  // 8 args: (neg_a, A, neg_b, B, c_mod, C, reuse_a, reuse_b)
  return __builtin_amdgcn_wmma_f32_16x16x4_f32(false, a, false, b, (short)0, c,
                                               false, false);
}

// ---------------------------------------------------------------------------
// Kernel 1: max-pooling matching.
//   out[i,b, col_base+m] = max_j  (Σ_h x1_i x2_j w_m^2) /
//                                 (||x1_i*w_m|| * ||x2_j*w_m||)  (eps-clamped)
// One block per batch element b (launched once per direction).
// Norms for ALL m computed up front as a WMMA GEMM of squared operands:
//   n1w[i,m]^2 = Σ_h X1[i,h]^2 * w[m,h]^2  -> (64x384)(X^2) @ (384x32)(W^2,
//   m padded to 32 with zero rows).
// ---------------------------------------------------------------------------
__global__ __launch_bounds__(256)
void mpm_kernel(const float* __restrict__ x1, const float* __restrict__ x2,
                const float* __restrict__ w, float* __restrict__ out,
                int col_base) {
  extern __shared__ float smem[];
  float* X1  = smem;                        // 64*388
  float* X2  = X1 + 64 * XSTRIDE;           // 64*388
  float* W   = X2 + 64 * XSTRIDE;           // 32*388 (rows 20..31 zeroed)
  float* RN1 = W + 32 * XSTRIDE;            // [m][row] 20*64 reciprocal norms
  float* RN2 = RN1 + MP_DIM * 64;           // 20*64
  float* RMX = RN2 + MP_DIM * 64;           // 64*4 row partial maxes

  const int b    = blockIdx.x;
  const int tid  = threadIdx.x;
  const int wid  = tid >> 5;
  const int lane = tid & 31;
  const int half = lane >> 4;
  const int l16  = lane & 15;

  // Stage X1_b, X2_b (64 rows x 384 cols, contiguous in H) as float4.
  for (int i = tid; i < 64 * 96; i += 256) {
    const int row = i / 96;
    const int c4  = (i % 96) * 4;
    const size_t src = ((size_t)(row * B_DIM + b)) * H_DIM + c4;
    *(float4*)(X1 + row * XSTRIDE + c4) = *(const float4*)(x1 + src);
    *(float4*)(X2 + row * XSTRIDE + c4) = *(const float4*)(x2 + src);
  }
  // Stage weight block (20 rows used, pad to 32 with zeros for the norm GEMM).
  for (int i = tid; i < 32 * 96; i += 256) {
    const int row = i / 96;
    const int c4  = (i % 96) * 4;
    float4 v = make_float4(0.f, 0.f, 0.f, 0.f);
    if (row < MP_DIM) v = *(const float4*)(w + row * H_DIM + c4);
    *(float4*)(W + row * XSTRIDE + c4) = v;
  }
  __syncthreads();

  // --- Norm GEMM: 16 tiles (mat in {X1,X2} x 4 row-tiles x 2 m-tiles). ---
  {
    const int mat = wid >> 2;               // 0 -> X1 norms, 1 -> X2 norms
    const int it  = wid & 3;
    const int i0  = it * 16;
    const float* X = mat ? X2 : X1;
    float* RN = mat ? RN2 : RN1;
    const int arow  = (i0 + l16) * XSTRIDE;
    const int brow0 = (l16) * XSTRIDE;        // m cols 0..15
    const int brow1 = (16 + l16) * XSTRIDE;   // m cols 16..31 (>=20 zeroed)
    v8f c0 = {};
    v8f c1 = {};
    for (int k = 0; k < H_DIM; k += 4) {
      const int ka = k + half * 2;
      v2f a, b0, b1;
      float ax = X[arow + ka],  ay = X[arow + ka + 1];
      float b0x = W[brow0 + ka], b0y = W[brow0 + ka + 1];
      float b1x = W[brow1 + ka], b1y = W[brow1 + ka + 1];
      a.x = ax * ax;   a.y = ay * ay;
      b0.x = b0x * b0x; b0.y = b0y * b0y;
      b1.x = b1x * b1x; b1.y = b1y * b1y;
      c0 = wmma_f32_4(a, b0, c0);
      c1 = wmma_f32_4(a, b1, c1);
    }
    for (int v = 0; v < 8; ++v) {
      const int row = i0 + v + half * 8;
      const int m0 = l16;            // always < 20
      const int m1 = 16 + l16;
      RN[m0 * 64 + row] = 1.0f / fmaxf(sqrtf(c0[v]), EPS_F);
      if (m1 < MP_DIM)
        RN[m1 * 64 + row] = 1.0f / fmaxf(sqrtf(c1[v]), EPS_F);
    }
  }
  __syncthreads();

  // --- Main loop: per perspective m, weighted 64x64 Gram + row max. ---
  const int tile0 = wid * 2;
  const int it  = tile0 >> 2;
  const int jt0 = tile0 & 3;
  const int jt1 = jt0 + 1;
  const int i0  = it * 16;
  const int arow  = (i0 + l16) * XSTRIDE;
  const int brow0 = (jt0 * 16 + l16) * XSTRIDE;
  const int brow1 = (jt1 * 16 + l16) * XSTRIDE;

  for (int m = 0; m < MP_DIM; ++m) {
    const float* WMr = W + m * XSTRIDE;
    v8f c0 = {};
    v8f c1 = {};
    for (int k = 0; k < H_DIM; k += 4) {
      const int ka = k + half * 2;
      const float w0 = WMr[ka], w1 = WMr[ka + 1];
      v2f a, b0, b1;
      a.x  = X1[arow + ka] * w0;      a.y  = X1[arow + ka + 1] * w1;
      b0.x = X2[brow0 + ka] * w0;     b0.y = X2[brow0 + ka + 1] * w1;
      b1.x = X2[brow1 + ka] * w0;     b1.y = X2[brow1 + ka + 1] * w1;
      c0 = wmma_f32_4(a, b0, c0);
      c1 = wmma_f32_4(a, b1, c1);
    }

    // cos = dot * rcp(n1[i]) * rcp(n2[j]); row-max over j within each tile.
    const float r20 = RN2[m * 64 + jt0 * 16 + l16];
    const float r21 = RN2[m * 64 + jt1 * 16 + l16];
    for (int v = 0; v < 8; ++v) {
      const int row = i0 + v + half * 8;
      const float r1 = RN1[m * 64 + row];
      float m0 = c0[v] * r1 * r20;
      float m1 = c1[v] * r1 * r21;
      for (int off = 8; off >= 1; off >>= 1) {
        m0 = fmaxf(m0, __shfl_xor(m0, off, 32));
        m1 = fmaxf(m1, __shfl_xor(m1, off, 32));
      }
      if (l16 == 0) { RMX[row * 4 + jt0] = m0; RMX[row * 4 + jt1] = m1; }
    }
    __syncthreads();

    if (tid < 64) {
      const float mx = fmaxf(fmaxf(RMX[tid * 4 + 0], RMX[tid * 4 + 1]),
                             fmaxf(RMX[tid * 4 + 2], RMX[tid * 4 + 3]));
      out[((size_t)(tid * B_DIM + b)) * 160 + col_base + m] = mx;
    }
    __syncthreads();
  }
}

// ---------------------------------------------------------------------------
// Kernel 2: cos_matrix + attentive prep.
//   cos[i,j] = (x1_i . x2_j) / (||x1_i|| * ||x2_j||)   (eps-clamped)
//   idx[i]   = argmax_j cos[i,j]          -> ws
//   meanA[i] = (cos[i,:] @ X2) / sum_j cos[i,j]  -> ws
// One block per batch element b (launched once per direction).
// ---------------------------------------------------------------------------
__global__ __launch_bounds__(256)
void attn_kernel(const float* __restrict__ x1, const float* __restrict__ x2,
                 float* __restrict__ meanA, int* __restrict__ idx_out) {
  extern __shared__ float smem[];
  float* X1   = smem;                      // 64*388
  float* X2   = X1 + 64 * XSTRIDE;         // 64*388
  float* COS  = X2 + 64 * XSTRIDE;         // 64*68
  float* RN1  = COS + 64 * CSTRIDE;        // 64
  float* RN2  = RN1 + 64;                  // 64
  float* RINV = RN2 + 64;                  // 64 (1 / row-sum of cos)
  float* PART = RINV + 64;                 // 256 norm partials

  const int b    = blockIdx.x;
  const int tid  = threadIdx.x;
  const int wid  = tid >> 5;
  const int lane = tid & 31;
  const int half = lane >> 4;
  const int l16  = lane & 15;

  for (int i = tid; i < 64 * 96; i += 256) {
    const int row = i / 96;
    const int c4  = (i % 96) * 4;
    const size_t src = ((size_t)(row * B_DIM + b)) * H_DIM + c4;
    *(float4*)(X1 + row * XSTRIDE + c4) = *(const float4*)(x1 + src);
    *(float4*)(X2 + row * XSTRIDE + c4) = *(const float4*)(x2 + src);
  }
  __syncthreads();

  // Unweighted row norms: 2 threads per row, float4 LDS reads.
  {
    const int r  = tid >> 1;               // 0..127 (64 X1 rows + 64 X2 rows)
    const int hh = tid & 1;
    const float* Xr = (r < 64 ? X1 + r * XSTRIDE : X2 + (r - 64) * XSTRIDE);
    float s = 0.f;
    for (int c4 = hh * 48; c4 < hh * 48 + 48; ++c4) {
      const float4 q = *(const float4*)(Xr + c4 * 4);
      s += q.x * q.x + q.y * q.y + q.z * q.z + q.w * q.w;
    }
    PART[tid] = s;
  }
  __syncthreads();
  if (tid < 128) {
    const float s = PART[2 * tid] + PART[2 * tid + 1];
    (tid < 64 ? RN1 : RN2)[tid & 63] = 1.0f / fmaxf(sqrtf(s), EPS_F);
  }
  __syncthreads();

  // 64x64 Gram matrix via WMMA, 2 tiles per wave.
  {
    const int tile0 = wid * 2;
    const int it  = tile0 >> 2;
    const int jt0 = tile0 & 3, jt1 = jt0 + 1;
    const int i0  = it * 16;
    const int arow  = (i0 + l16) * XSTRIDE;
    const int brow0 = (jt0 * 16 + l16) * XSTRIDE;
    const int brow1 = (jt1 * 16 + l16) * XSTRIDE;
    v8f c0 = {};
    v8f c1 = {};
    for (int k = 0; k < H_DIM; k += 4) {
      const int ka = k + half * 2;
      v2f a, b0, b1;
      a.x  = X1[arow + ka];   a.y  = X1[arow + ka + 1];
      b0.x = X2[brow0 + ka];  b0.y = X2[brow0 + ka + 1];
      b1.x = X2[brow1 + ka];  b1.y = X2[brow1 + ka + 1];
      c0 = wmma_f32_4(a, b0, c0);
      c1 = wmma_f32_4(a, b1, c1);
    }
    const float r20 = RN2[jt0 * 16 + l16];
    const float r21 = RN2[jt1 * 16 + l16];
    for (int v = 0; v < 8; ++v) {
      const int row = i0 + v + half * 8;
      const float r1 = RN1[row];
      COS[row * CSTRIDE + jt0 * 16 + l16] = c0[v] * r1 * r20;
      COS[row * CSTRIDE + jt1 * 16 + l16] = c1[v] * r1 * r21;
    }
  }
  __syncthreads();

  // Row sums (for attentive mean) + argmax (for max-attentive gather).
  if (tid < 64) {
    float s = 0.f, best = -INFINITY;
    int bi = 0;
    for (int j = 0; j < 64; ++j) {
      const float v = COS[tid * CSTRIDE + j];
      s += v;
      if (v > best) { best = v; bi = j; }
    }
    RINV[tid] = 1.0f / s;
    idx_out[tid * B_DIM + b] = bi;
  }
  __syncthreads();

  // meanA (64x384) = COS (64x64) @ X2 (64x384), scaled by RINV; K=64.
  for (int t = wid; t < 96; t += 8) {
    const int i0 = (t / 24) * 16;
    const int h0 = (t % 24) * 16;
    v8f c = {};
    for (int k = 0; k < 64; k += 4) {
      const int ka = k + half * 2;
      v2f a, bb;
      a.x  = COS[(i0 + l16) * CSTRIDE + ka];
      a.y  = COS[(i0 + l16) * CSTRIDE + ka + 1];
      bb.x = X2[ka * XSTRIDE + h0 + l16];
      bb.y = X2[(ka + 1) * XSTRIDE + h0 + l16];
      c = wmma_f32_4(a, bb, c);
    }
    for (int v = 0; v < 8; ++v) {
      const int row = i0 + v + half * 8;
      meanA[((size_t)(row * B_DIM + b)) * H_DIM + h0 + l16] = c[v] * RINV[row];
    }
  }
}

// ---------------------------------------------------------------------------
// Kernel 3: full / attentive / max-attentive matching finals.
// One block per (t,b) (launched once per direction). 60 weighted-cosine
// reductions (3 types x 20 perspectives) over H=384, one per wave-task.
// ---------------------------------------------------------------------------
__global__ __launch_bounds__(256)
void final_kernel(const float* __restrict__ x1, const float* __restrict__ x2,
                  const float* __restrict__ meanA, const int* __restrict__ idx,
                  const float* __restrict__ w_fm, const float* __restrict__ w_am,
                  const float* __restrict__ w_mam, float* __restrict__ out,
                  int col_base) {
  __shared__ float U[H_DIM], VFM[H_DIM], VAM[H_DIM], VMAM[H_DIM];
  const int bid = blockIdx.x;
  const int t   = bid / B_DIM;
  const int b   = bid % B_DIM;
  const int tid = threadIdx.x;

  const int g = idx[t * B_DIM + b];  // argmax_j cos[t,j]
  for (int h = tid; h < H_DIM; h += 256) {
    U[h]    = x1[((size_t)(t * B_DIM + b)) * H_DIM + h];
    VFM[h]  = x2[((size_t)((T_DIM - 1) * B_DIM + b)) * H_DIM + h];
    VAM[h]  = meanA[((size_t)(t * B_DIM + b)) * H_DIM + h];
    VMAM[h] = x2[((size_t)(g * B_DIM + b)) * H_DIM + h];
  }
  __syncthreads();

  const int wid = tid >> 5, lane = tid & 31;
  for (int task = wid; task < 60; task += 8) {
    const int type = task / 20;
    const int m    = task % 20;
    const float* wp = (type == 0 ? w_fm : type == 1 ? w_am : w_mam) + m * H_DIM;
    const float* V  = (type == 0 ? VFM : type == 1 ? VAM : VMAM);
    float dot = 0.f, nu = 0.f, nv = 0.f;
    for (int h = lane; h < H_DIM; h += 32) {
      const float wv = wp[h];
      const float w2 = wv * wv;
      const float uu = U[h], vv = V[h];
      dot += uu * vv * w2;
      nu  += uu * uu * w2;
      nv  += vv * vv * w2;
    }
    for (int off = 16; off >= 1; off >>= 1) {
      dot += __shfl_xor(dot, off, 32);
      nu  += __shfl_xor(nu,  off, 32);
      nv  += __shfl_xor(nv,  off, 32);
    }
    if (lane == 0) {
      const float cs = dot / (fmaxf(sqrtf(nu), EPS_F) * fmaxf(sqrtf(nv), EPS_F));
      const int col = (type == 0 ? 0 : type == 1 ? 80 : 120) + col_base + m;
      out[((size_t)(t * B_DIM + b)) * 160 + col] = cs;
    }
  }
}

// ---------------------------------------------------------------------------
extern "C" void kernel_launch(void* const* d_in, const int* in_sizes, int n_in,
                              void* d_out, int out_size, void* d_ws, size_t ws_size,
                              hipStream_t stream) {
  const float* p_fw = (const float*)d_in[0];
  const float* p_bw = (const float*)d_in[1];
  const float* h_fw = (const float*)d_in[2];
  const float* h_bw = (const float*)d_in[3];
  const float* w_fm_fw  = (const float*)d_in[4];
  const float* w_fm_bw  = (const float*)d_in[5];
  const float* w_mpm_fw = (const float*)d_in[6];
  const float* w_mpm_bw = (const float*)d_in[7];
  const float* w_am_fw  = (const float*)d_in[8];
  const float* w_am_bw  = (const float*)d_in[9];
  const float* w_mam_fw = (const float*)d_in[10];
  const float* w_mam_bw = (const float*)d_in[11];
  float* out = (float*)d_out;

  // Workspace: meanA[2][T][B][H] floats (25.2 MB) + idx[2][T][B] ints (64 KB).
  const size_t meanA_elems = (size_t)T_DIM * B_DIM * H_DIM;
  float* meanA = (float*)d_ws;
  int* idxws = (int*)((char*)d_ws + 2 * meanA_elems * sizeof(float));

  const size_t lds1 = (size_t)(2 * 64 * XSTRIDE + 32 * XSTRIDE +
                               2 * MP_DIM * 64 + 64 * 4) * sizeof(float);
  const size_t lds2 = (size_t)(2 * 64 * XSTRIDE + 64 * CSTRIDE + 3 * 64 + 256) *
                      sizeof(float);

  mpm_kernel<<<B_DIM, 256, lds1, stream>>>(p_fw, h_fw, w_mpm_fw, out, 40);
  mpm_kernel<<<B_DIM, 256, lds1, stream>>>(p_bw, h_bw, w_mpm_bw, out, 60);

  attn_kernel<<<B_DIM, 256, lds2, stream>>>(p_fw, h_fw, meanA, idxws);
  attn_kernel<<<B_DIM, 256, lds2, stream>>>(p_bw, h_bw, meanA + meanA_elems,
                                            idxws + T_DIM * B_DIM);

  final_kernel<<<T_DIM * B_DIM, 256, 0, stream>>>(
      p_fw, h_fw, meanA, idxws, w_fm_fw, w_am_fw, w_mam_fw, out, 0);
  final_kernel<<<T_DIM * B_DIM, 256, 0, stream>>>(
      p_bw, h_bw, meanA + meanA_elems, idxws + T_DIM * B_DIM,
      w_fm_bw, w_am_bw, w_mam_bw, out, 20);
}